// SpatialTransformer_2499670966794
// MI455X (gfx1250) — compile-verified
//
#include <hip/hip_runtime.h>
#include <stdint.h>

// 3D spatial-transformer warp (trilinear gather) for MI455X / gfx1250.
// Memory-bound: ~315 MB of traffic -> ~13.5us at 23.3 TB/s. No matrix
// structure (0.5 FLOP/byte), so WMMA is inapplicable; the CDNA5-specific
// path used is the async global->LDS DMA (global_load_async_to_lds_b128 +
// s_wait_asynccnt) for the streaming flow reads, LDS b128 reads, and
// nontemporal b128 stores for the outputs.
//
// v2: compute (z,y,x) decomposition once per thread instead of per voxel.
// A thread's 4 voxels are consecutive and 4-aligned; W=160 and H*W=30720
// are multiples of 4, so the group never crosses a row/slice boundary and
// only x increments within the group. Removes 6 of 8 magic-number division
// sequences per thread (~1/3 of the VALU work).

typedef float v4f __attribute__((ext_vector_type(4)));

#define B_ 2
#define D_ 160
#define H_ 192
#define W_ 160
#define N_ (D_ * H_ * W_)        /* 4,915,200 voxels per batch         */
#define HW_ (H_ * W_)            /* 30,720                             */
#define TILE_ 1024               /* voxels per block (divides N_)      */
#define TPB_ 256                 /* 8 wave32s per block                */
#define TILES_PER_B (N_ / TILE_) /* 4800                               */

__device__ __forceinline__ int iclamp(int v, int lo, int hi) {
    v = v < lo ? lo : v;
    return v > hi ? hi : v;
}

struct Axis { float w0, w1; int i0, i1; };

// Per-axis corner weights with the OOB mask folded into the weight and
// indices clamped, so the 8 gather loads are always in-bounds and the
// lanes never diverge.
__device__ __forceinline__ Axis mkaxis(float q, int S) {
    float f = floorf(q);
    int   i = (int)f;
    float a = q - f;               // in [0,1)
    Axis A;
    A.w0 = (i     >= 0 && i     < S) ? (1.0f - a) : 0.0f;
    A.w1 = (i + 1 >= 0 && i + 1 < S) ? a          : 0.0f;
    A.i0 = iclamp(i,     0, S - 1);
    A.i1 = iclamp(i + 1, 0, S - 1);
    return A;
}

__global__ __launch_bounds__(TPB_)
void warp3d_kernel(const float* __restrict__ src,
                   const float* __restrict__ flow,
                   float* __restrict__ out) {
    __shared__ __align__(16) float ftile[3 * TILE_];   // 12 KB / block

    const int b    = blockIdx.x / TILES_PER_B;
    const int base = (blockIdx.x % TILES_PER_B) * TILE_;
    const int t    = threadIdx.x;

    // Workgroup-relative LDS byte offset of ftile (low 32 bits of the flat
    // address of a __shared__ object are the LDS offset on gfx1250).
    const unsigned ldsBase = (unsigned)(uintptr_t)&ftile[0];
    const unsigned long long flowAddr = (unsigned long long)(uintptr_t)flow;

    // --- Async DMA: stage this tile's three flow channels into LDS -------
    // Each lane moves 16B per channel; 256 lanes x 3 = 12 KB.
#pragma unroll
    for (int c = 0; c < 3; ++c) {
        unsigned gOff = (unsigned)(((unsigned)(b * 3 + c) * (unsigned)N_ +
                                    (unsigned)(base + t * 4)) * 4u);
        unsigned lOff = ldsBase + (unsigned)((c * TILE_ + t * 4) * 4);
        asm volatile("global_load_async_to_lds_b128 %0, %1, %2"
                     :: "v"(lOff), "v"(gOff), "s"(flowAddr)
                     : "memory");
    }
    asm volatile("s_wait_asynccnt 0x0" ::: "memory");
    __syncthreads();

    // LDS -> registers (b128 reads); reused for both compute and the
    // flow pass-through output (read flow from HBM exactly once).
    v4f fz = ((const v4f*)&ftile[0 * TILE_])[t];
    v4f fy = ((const v4f*)&ftile[1 * TILE_])[t];
    v4f fx = ((const v4f*)&ftile[2 * TILE_])[t];

    const float* __restrict__ sb = src + (size_t)b * N_;

    const float sclD = (float)D_ / (float)(D_ - 1);   // 160/159
    const float sclH = (float)H_ / (float)(H_ - 1);   // 192/191
    const float sclW = (float)W_ / (float)(W_ - 1);   // 160/159

    // Decompose the first voxel index once; the 4 voxels of this thread
    // share (z, y) and differ only by x (see header comment for proof).
    const int v0 = base + t * 4;
    const int z  = v0 / HW_;
    const int r  = v0 - z * HW_;
    const int y  = r / W_;
    const int x0 = r - y * W_;
    const float zf = (float)z;
    const float yf = (float)y;

    v4f res;
#pragma unroll
    for (int k = 0; k < 4; ++k) {
        // q = (coord + flow) * s/(s-1) - 0.5  (algebraic reduction of the
        // reference's normalize/denormalize round-trip)
        const float qz = (zf + fz[k]) * sclD - 0.5f;
        const float qy = (yf + fy[k]) * sclH - 0.5f;
        const float qx = ((float)(x0 + k) + fx[k]) * sclW - 0.5f;

        const Axis Z = mkaxis(qz, D_);
        const Axis Y = mkaxis(qy, H_);
        const Axis X = mkaxis(qx, W_);

        const int zy00 = Z.i0 * HW_ + Y.i0 * W_;
        const int zy01 = Z.i0 * HW_ + Y.i1 * W_;
        const int zy10 = Z.i1 * HW_ + Y.i0 * W_;
        const int zy11 = Z.i1 * HW_ + Y.i1 * W_;

        const float v000 = sb[zy00 + X.i0], v001 = sb[zy00 + X.i1];
        const float v010 = sb[zy01 + X.i0], v011 = sb[zy01 + X.i1];
        const float v100 = sb[zy10 + X.i0], v101 = sb[zy10 + X.i1];
        const float v110 = sb[zy11 + X.i0], v111 = sb[zy11 + X.i1];

        res[k] = Z.w0 * (Y.w0 * (X.w0 * v000 + X.w1 * v001) +
                         Y.w1 * (X.w0 * v010 + X.w1 * v011)) +
                 Z.w1 * (Y.w0 * (X.w0 * v100 + X.w1 * v101) +
                         Y.w1 * (X.w0 * v110 + X.w1 * v111));
    }

    // --- Stores (streaming, no reuse -> nontemporal b128) ----------------
    // Output layout: warped (B,1,D,H,W) followed by flow (B,3,D,H,W).
    v4f* wOut = (v4f*)(out + (size_t)b * N_ + base);
    __builtin_nontemporal_store(res, wOut + t);

    float* fOut = out + (size_t)2 * N_;
    __builtin_nontemporal_store(fz, (v4f*)(fOut + (size_t)(b * 3 + 0) * N_ + base) + t);
    __builtin_nontemporal_store(fy, (v4f*)(fOut + (size_t)(b * 3 + 1) * N_ + base) + t);
    __builtin_nontemporal_store(fx, (v4f*)(fOut + (size_t)(b * 3 + 2) * N_ + base) + t);
}

extern "C" void kernel_launch(void* const* d_in, const int* in_sizes, int n_in,
                              void* d_out, int out_size, void* d_ws, size_t ws_size,
                              hipStream_t stream) {
    (void)in_sizes; (void)n_in; (void)out_size; (void)d_ws; (void)ws_size;
    const float* src  = (const float*)d_in[0];   // (2,1,160,192,160) f32
    const float* flow = (const float*)d_in[1];   // (2,3,160,192,160) f32
    float* out = (float*)d_out;                  // warped ++ flow, f32

    const int grid = B_ * TILES_PER_B;           // 9600 blocks
    warp3d_kernel<<<grid, TPB_, 0, stream>>>(src, flow, out);
}